// DCN_v2_14645838480160
// MI455X (gfx1250) — compile-verified
//
#include <hip/hip_runtime.h>
#include <hip/hip_bf16.h>
#include <math.h>

// DCN v2 for MI455X (gfx1250, wave32).
// B=8, C=64, H=W=96, O=64, K=3, pad=1, stride=1.
// d_out layout (fp32, concatenated): offset[8,18,96,96] | mask[8,9,96,96] | y[8,64,96,96]
//
// Both convolutions are K=576 GEMMs fed from an LDS im2col tile:
//   kernel 1: plain shifted-window cols, A = w_off (27 rows, padded to 32)
//   kernel 2: bilinear deformable cols,  A = w_core (64 rows)
// GEMM runs on V_WMMA_F32_16X16X4_F32 (fp32-exact, matches reference numerics).
// Weight staging uses GLOBAL_LOAD_ASYNC_TO_LDS_B128 (ASYNCcnt).

#define Bsz 8
#define Cch 64
#define Hh  96
#define Ww  96
#define HW  (Hh*Ww)
#define K2  9
#define KR  (Cch*K2)     // 576 reduction length
#define OCH 64

#define WSM_STRIDE 580   // weight LDS row stride (576+4): lanes 0..15 hit banks 4*m -> conflict-free
#define COL_STRIDE 72    // cols LDS row stride (64+8): rows r,r+2 differ by 144 (mod 64 = 16) -> disjoint bank sets
#define NPIX 64          // pixels per workgroup (four 16-wide WMMA N tiles)

typedef __attribute__((ext_vector_type(2)))  float  v2f;
typedef __attribute__((ext_vector_type(8)))  float  v8f;

// Async global->LDS copy, 16B per lane (GLOBAL_LOAD_ASYNC_TO_LDS_B128, ASYNCcnt).
// lds_off is the hardware LDS byte address (== byte offset from dynamic-LDS base).
__device__ __forceinline__ void async_g2l_b128(unsigned lds_off, const float* gsrc) {
  asm volatile("global_load_async_to_lds_b128 %0, %1, off"
               :: "v"(lds_off), "v"((unsigned long long)(uintptr_t)gsrc)
               : "memory");
}
__device__ __forceinline__ void async_wait_all() {
  asm volatile("s_wait_asynccnt 0" ::: "memory");
}

// Fast sigmoid: v_exp_f32 + v_rcp_f32 (plenty of precision for a gating mask).
__device__ __forceinline__ float fast_sigmoid(float r) {
  return __builtin_amdgcn_rcpf(1.0f + __expf(-r));
}

// ---------------------------------------------------------------------------
// Kernel 1: 3x3 offset conv (27 ch) via WMMA + sigmoid on mask channels.
// 256 threads. LDS: w_off [32][580] (rows 27..31 zero) + cols [576][72].
// Waves: Mi = wave&1 (16 out-ch), Nj = wave>>1 (16 pixels). 2x4 = 8 waves.
// ---------------------------------------------------------------------------
__global__ __launch_bounds__(256) void offset_conv_kernel(
    const float* __restrict__ x, const float* __restrict__ w_off,
    const float* __restrict__ b_off, float* __restrict__ out) {
  extern __shared__ float smem[];
  float* wofs = smem;                        // [32][WSM_STRIDE]
  float* cols = smem + 32 * WSM_STRIDE;      // [KR][COL_STRIDE]

  const int tid = threadIdx.x;
  const int blk = blockIdx.x;
  const int b = blk / (HW / NPIX);
  const int hwbase = (blk % (HW / NPIX)) * NPIX;

  // ---- Phase 0a: async-stage w_off [27][576] into padded LDS rows
  for (int i4 = tid; i4 < 27 * (KR / 4); i4 += 256) {
    const int o = i4 / (KR / 4), rq = i4 % (KR / 4);
    async_g2l_b128((unsigned)((o * WSM_STRIDE + rq * 4) * 4), w_off + i4 * 4);
  }
  // ---- Phase 0b: zero pad rows 27..31
  for (int i = tid; i < 5 * WSM_STRIDE; i += 256)
    wofs[27 * WSM_STRIDE + i] = 0.0f;

  // ---- Phase 1: plain im2col (zero-padded 3x3 window) -> cols
  const float* xb = x + (size_t)b * Cch * HW;
  for (int p = tid; p < NPIX * K2; p += 256) {
    const int pix = p % NPIX;          // lanes 0..31 -> consecutive pixels, same tap
    const int k   = p / NPIX;
    const int hw  = hwbase + pix;
    const int h = hw / Ww, w = hw % Ww;
    const int hy = h - 1 + (k / 3);
    const int wx = w - 1 + (k % 3);
    const bool inr = (hy >= 0) & (hy < Hh) & (wx >= 0) & (wx < Ww);
    const int idx = (inr ? hy : 0) * Ww + (inr ? wx : 0);
    float* cdst = cols + k * COL_STRIDE + pix;
#pragma unroll 4
    for (int c = 0; c < Cch; ++c) {
      const float v = inr ? xb[c * HW + idx] : 0.0f;
      cdst[c * K2 * COL_STRIDE] = v;
    }
  }
  async_wait_all();
  __syncthreads();

  // ---- Phase 2: WMMA GEMM, D[16x16] += A[16x4]*B[4x16], 144 K-steps
  const int wave = tid >> 5, lane = tid & 31;
  const int Mi = wave & 1, Nj = wave >> 1;
  const int mrow = Mi * 16 + (lane & 15);
  const int ncol = Nj * 16 + (lane & 15);
  const int koff = (lane < 16) ? 0 : 2;
  const float* arow = wofs + mrow * WSM_STRIDE + koff;

  v8f acc = {0.f, 0.f, 0.f, 0.f, 0.f, 0.f, 0.f, 0.f};
#pragma unroll 4
  for (int kk = 0; kk < KR / 4; ++kk) {
    v2f a, bf;
    a.x  = arow[kk * 4];
    a.y  = arow[kk * 4 + 1];
    bf.x = cols[(kk * 4 + koff    ) * COL_STRIDE + ncol];
    bf.y = cols[(kk * 4 + koff + 1) * COL_STRIDE + ncol];
    acc = __builtin_amdgcn_wmma_f32_16x16x4_f32(
        false, a, false, bf, (short)0, acc, false, false);
  }

  // ---- Store: D VGPR v -> M = Mi*16 + v (+8 hi half), N = lane%16
  float* offp = out;                          // [B][18][HW]
  float* mskp = out + (size_t)Bsz * 18 * HW;  // [B][9][HW]
  const int mofs = (lane < 16) ? 0 : 8;
  const int hw = hwbase + ncol;
#pragma unroll
  for (int v = 0; v < 8; ++v) {
    const int o = Mi * 16 + v + mofs;
    const float r = acc[v] + b_off[min(o, 26)];
    if (o < 18)       offp[(size_t)(b * 18 + o) * HW + hw] = r;
    else if (o < 27)  mskp[(size_t)(b * 9 + (o - 18)) * HW + hw] = fast_sigmoid(r);
  }
}

// ---------------------------------------------------------------------------
// Kernel 2: bilinear deformable gather into LDS + WMMA GEMM with w_core.
// 256 threads. LDS: w_core [64][580] + cols [576][72] = 314 KB (WGP has 320 KB).
// Waves: Mi = wave&3; each wave runs TWO N tiles (A-fragment reuse).
// ---------------------------------------------------------------------------
__global__ __launch_bounds__(256) void dcn_core_kernel(
    const float* __restrict__ x, const float* __restrict__ w_core,
    const float* __restrict__ outbuf, float* __restrict__ y) {
  extern __shared__ float smem[];
  float* wsm  = smem;                        // [OCH][WSM_STRIDE]
  float* cols = smem + OCH * WSM_STRIDE;     // [KR][COL_STRIDE], row r = c*9 + k

  const int tid = threadIdx.x;
  const int blk = blockIdx.x;
  const int b = blk / (HW / NPIX);
  const int hwbase = (blk % (HW / NPIX)) * NPIX;

  // ---- Phase 0: async-stage w_core [64][576] into padded LDS rows
  for (int i4 = tid; i4 < OCH * (KR / 4); i4 += 256) {
    const int o = i4 / (KR / 4), rq = i4 % (KR / 4);
    async_g2l_b128((unsigned)((o * WSM_STRIDE + rq * 4) * 4), w_core + i4 * 4);
  }

  // ---- Phase 1: bilinear deformable gather -> cols
  const float* offp = outbuf;
  const float* mskp = outbuf + (size_t)Bsz * 18 * HW;
  const float* xb = x + (size_t)b * Cch * HW;
  for (int p = tid; p < NPIX * K2; p += 256) {
    const int pix = p % NPIX;          // lanes 0..31 -> consecutive pixels, same tap
    const int k   = p / NPIX;
    const int hw  = hwbase + pix;
    const int h = hw / Ww, w = hw % Ww;

    const float o1v = offp[(size_t)(b * 18 + k) * HW + hw];
    const float o2v = offp[(size_t)(b * 18 + 9 + k) * HW + hw];
    const float mv  = mskp[(size_t)(b * 9 + k) * HW + hw];

    const float py = (float)(h - 1 + (k / 3)) + o1v;
    const float px = (float)(w - 1 + (k % 3)) + o2v;
    const float y0f = floorf(py), x0f = floorf(px);
    const float ly = py - y0f, lx = px - x0f;
    const int y0 = (int)y0f, x0 = (int)x0f;
    const int y1 = y0 + 1,  x1 = x0 + 1;
    const float vy0 = (y0 >= 0 && y0 < Hh) ? 1.0f : 0.0f;
    const float vy1 = (y1 >= 0 && y1 < Hh) ? 1.0f : 0.0f;
    const float vx0 = (x0 >= 0 && x0 < Ww) ? 1.0f : 0.0f;
    const float vx1 = (x1 >= 0 && x1 < Ww) ? 1.0f : 0.0f;
    const int y0c = min(max(y0, 0), Hh - 1), y1c = min(max(y1, 0), Hh - 1);
    const int x0c = min(max(x0, 0), Ww - 1), x1c = min(max(x1, 0), Ww - 1);
    const float w00 = mv * (1.0f - ly) * (1.0f - lx) * vy0 * vx0;
    const float w01 = mv * (1.0f - ly) * lx          * vy0 * vx1;
    const float w10 = mv * ly          * (1.0f - lx) * vy1 * vx0;
    const float w11 = mv * ly          * lx          * vy1 * vx1;
    const int i00 = y0c * Ww + x0c, i01 = y0c * Ww + x1c;
    const int i10 = y1c * Ww + x0c, i11 = y1c * Ww + x1c;

    float* cdst = cols + k * COL_STRIDE + pix;   // row (c*9 + k)
#pragma unroll 4
    for (int c = 0; c < Cch; ++c) {
      const float* xc = xb + c * HW;
      const float v = fmaf(w00, xc[i00],
                      fmaf(w01, xc[i01],
                      fmaf(w10, xc[i10], w11 * xc[i11])));
      cdst[c * K2 * COL_STRIDE] = v;
    }
  }
  async_wait_all();
  __syncthreads();

  // ---- Phase 2: WMMA GEMM, one M tile x two N tiles per wave (A reuse)
  const int wave = tid >> 5, lane = tid & 31;
  const int Mi = wave & 3;
  const int NjB = (wave >> 2) * 2;           // N tiles NjB and NjB+1
  const int mrow = Mi * 16 + (lane & 15);
  const int ncol0 = NjB * 16 + (lane & 15);
  const int koff = (lane < 16) ? 0 : 2;
  const float* arow = wsm + mrow * WSM_STRIDE + koff;

  v8f acc0 = {0.f, 0.f, 0.f, 0.f, 0.f, 0.f, 0.f, 0.f};
  v8f acc1 = acc0;
#pragma unroll 2
  for (int kk = 0; kk < KR / 4; ++kk) {
    v2f a, b0, b1;
    a.x  = arow[kk * 4];
    a.y  = arow[kk * 4 + 1];
    const float* crow0 = cols + (kk * 4 + koff) * COL_STRIDE + ncol0;
    b0.x = crow0[0];
    b0.y = crow0[COL_STRIDE];
    b1.x = crow0[16];
    b1.y = crow0[COL_STRIDE + 16];
    acc0 = __builtin_amdgcn_wmma_f32_16x16x4_f32(
        false, a, false, b0, (short)0, acc0, false, false);
    acc1 = __builtin_amdgcn_wmma_f32_16x16x4_f32(
        false, a, false, b1, (short)0, acc1, false, false);
  }

  // ---- Store: D VGPR v -> M = Mi*16 + v (+8 hi half), N = lane%16
  float* yb = y + (size_t)b * OCH * HW + hwbase;
  const int mofs = (lane < 16) ? 0 : 8;
#pragma unroll
  for (int v = 0; v < 8; ++v) {
    const int o = Mi * 16 + v + mofs;
    yb[(size_t)o * HW + ncol0]      = acc0[v];
    yb[(size_t)o * HW + ncol0 + 16] = acc1[v];
  }
}

// ---------------------------------------------------------------------------
extern "C" void kernel_launch(void* const* d_in, const int* in_sizes, int n_in,
                              void* d_out, int out_size, void* d_ws, size_t ws_size,
                              hipStream_t stream) {
  const float* x      = (const float*)d_in[0];
  const float* w_off  = (const float*)d_in[1];
  const float* b_off  = (const float*)d_in[2];
  const float* w_core = (const float*)d_in[3];
  float* out = (float*)d_out;
  float* yout = out + (size_t)Bsz * (18 + 9) * HW;  // y section

  const int nblk = Bsz * (HW / NPIX);  // 1152

  const size_t smem1 = (size_t)(32 * WSM_STRIDE + KR * COL_STRIDE) * sizeof(float);
  (void)hipFuncSetAttribute((const void*)offset_conv_kernel,
                            hipFuncAttributeMaxDynamicSharedMemorySize, (int)smem1);
  offset_conv_kernel<<<dim3(nblk), dim3(256), smem1, stream>>>(x, w_off, b_off, out);

  const size_t smem2 = (size_t)(OCH * WSM_STRIDE + KR * COL_STRIDE) * sizeof(float);
  (void)hipFuncSetAttribute((const void*)dcn_core_kernel,
                            hipFuncAttributeMaxDynamicSharedMemorySize, (int)smem2);
  dcn_core_kernel<<<dim3(nblk), dim3(256), smem2, stream>>>(x, w_core, out, yout);
}